// GraphConvolution_47940424958090
// MI455X (gfx1250) — compile-verified
//
#include <hip/hip_runtime.h>

typedef __attribute__((ext_vector_type(2))) float v2f;
typedef __attribute__((ext_vector_type(8))) float v8f;

#define D_FEAT 64

// ---------------------------------------------------------------------------
// Kernel 1: support = h_v @ W using V_WMMA_F32_16X16X4_F32.
// One wave computes one 16x16 tile of the output. K=64 -> 16 WMMA steps.
// A layout (16x4 f32, 2 VGPRs): lane L holds row M=L&15; VGPR0/1 hold
//   K = 2*(L>>4), 2*(L>>4)+1  -> one float2 load from row-major h_v.
// B layout (4x16 f32, 2 VGPRs): lane L holds col N=L&15; halves split K same
//   as A -> two strided scalar loads from row-major W.
// C/D layout (16x16 f32, 8 VGPRs): acc[i] = row (i + 8*(L>>4)), col L&15.
// ---------------------------------------------------------------------------
__global__ void gcn_gemm_wmma_f32(const float* __restrict__ A,   // [n_rows, 64]
                                  const float* __restrict__ W,   // [64, 64]
                                  float* __restrict__ S,         // [n_rows, 64]
                                  int n_rows) {
  const int lane  = threadIdx.x & 31;
  const int wid   = (blockIdx.x * blockDim.x + threadIdx.x) >> 5;
  const int tiles_n = D_FEAT / 16;                 // 4 column tiles
  const int tile_m  = wid / tiles_n;
  const int tile_n  = wid % tiles_n;
  if (tile_m * 16 >= n_rows) return;               // wave-uniform: EXEC stays full

  const int half = lane >> 4;                      // 0 or 1
  const int ln   = lane & 15;                      // A row / B,C column within tile
  const int row  = tile_m * 16 + ln;               // A row in memory
  const int col  = tile_n * 16 + ln;               // B/C/D column in memory

  v8f acc = {};
#pragma unroll
  for (int kk = 0; kk < D_FEAT; kk += 4) {
    const int ka = kk + 2 * half;
    v2f a = *(const v2f*)(A + (size_t)row * D_FEAT + ka);
    v2f b;
    b.x = W[(ka + 0) * D_FEAT + col];
    b.y = W[(ka + 1) * D_FEAT + col];
    // (neg_a, A, neg_b, B, c_mod, C, reuse_a, reuse_b)
    acc = __builtin_amdgcn_wmma_f32_16x16x4_f32(false, a, false, b,
                                                (short)0, acc, false, false);
  }

#pragma unroll
  for (int i = 0; i < 8; ++i) {
    const int out_row = tile_m * 16 + i + 8 * half;
    S[(size_t)out_row * D_FEAT + col] = acc[i];
  }
}

// ---------------------------------------------------------------------------
// Kernel 2: out[i][j] = bias[j]  (initialize accumulator target)
// ---------------------------------------------------------------------------
__global__ void gcn_init_bias(float* __restrict__ out,
                              const float* __restrict__ bias, int total) {
  int i = blockIdx.x * blockDim.x + threadIdx.x;
  if (i < total) out[i] = bias[i & (D_FEAT - 1)];
}

// ---------------------------------------------------------------------------
// Kernel 3: scatter SpMM. One wave per edge per iteration; each lane covers
// feature columns (2*lane, 2*lane+1): coalesced float2 gather of
// support[cols[e]] (L2-resident, 25.6MB << 192MB L2) and two f32 atomic adds
// into out[rows[e]] (also L2-resident -> atomics resolve in L2).
// ---------------------------------------------------------------------------
__global__ void gcn_spmm_scatter(const int*   __restrict__ rows,
                                 const int*   __restrict__ cols,
                                 const float* __restrict__ vals,
                                 const float* __restrict__ S,
                                 float*       __restrict__ out,
                                 int n_edges) {
  const int lane    = threadIdx.x & 31;
  const int wid     = (blockIdx.x * blockDim.x + threadIdx.x) >> 5;
  const int n_waves = (gridDim.x * blockDim.x) >> 5;

  for (int e = wid; e < n_edges; e += n_waves) {
    const int   r = rows[e];
    const int   c = cols[e];
    const float v = vals[e];
    v2f s = *(const v2f*)(S + (size_t)c * D_FEAT + 2 * lane);
    float* dst = out + (size_t)r * D_FEAT + 2 * lane;
    atomicAdd(dst + 0, v * s.x);
    atomicAdd(dst + 1, v * s.y);
  }
}

// ---------------------------------------------------------------------------
extern "C" void kernel_launch(void* const* d_in, const int* in_sizes, int n_in,
                              void* d_out, int out_size, void* d_ws, size_t ws_size,
                              hipStream_t stream) {
  const float* h_v   = (const float*)d_in[0];   // [N, 64]
  const int*   erows = (const int*)  d_in[1];   // [E]
  const int*   ecols = (const int*)  d_in[2];   // [E]
  const float* evals = (const float*)d_in[3];   // [E]
  const float* W     = (const float*)d_in[4];   // [64, 64]
  const float* bias  = (const float*)d_in[5];   // [64]
  float*       out   = (float*)d_out;           // [N, 64]
  float*       support = (float*)d_ws;          // [N, 64] scratch

  const int n_nodes = in_sizes[0] / D_FEAT;
  const int n_edges = in_sizes[1];

  // GEMM: one wave per 16x16 tile; 256-thread blocks = 8 waves.
  const int n_tiles = ((n_nodes + 15) / 16) * (D_FEAT / 16);
  const int gemm_blocks = (n_tiles + 7) / 8;
  gcn_gemm_wmma_f32<<<gemm_blocks, 256, 0, stream>>>(h_v, W, support, n_nodes);

  // out = bias (broadcast)
  const int total = n_nodes * D_FEAT;
  gcn_init_bias<<<(total + 255) / 256, 256, 0, stream>>>(out, bias, total);

  // Scatter-add: enough waves to saturate, grid-stride over edges.
  gcn_spmm_scatter<<<2048, 256, 0, stream>>>(erows, ecols, evals, support,
                                             out, n_edges);
}